// DeformableCrossAttentionModule_18433999634673
// MI455X (gfx1250) — compile-verified
//
#include <hip/hip_runtime.h>
#include <math.h>

// Problem constants (match reference)
#define Bc   4
#define Cc   256
#define Hc   64
#define Wc   64
#define Nn   9
#define OUTC 256
#define Kk   (Cc * Nn)        // 2304
#define PIX  (Hc * Wc)        // 4096
#define Mtot (Bc * PIX)       // 16384
#define KC   8                // channels per staged chunk
#define KROWS (KC * Nn)       // 72 k-rows per chunk
#define ASTR 33               // LDS row stride for A (bank-conflict pad)

typedef __attribute__((ext_vector_type(2))) float v2f;
typedef __attribute__((ext_vector_type(8))) float v8f;

// ---------------------------------------------------------------------------
// Kernel 1: 3x3 convs producing offset (18 ch) and sigmoid(mod) (9 ch).
// grid: (PIX*Bc/256, 27), block 256. blockIdx.y = output channel j.
// ---------------------------------------------------------------------------
__global__ void dca_conv_kernel(const float* __restrict__ query,
                                const float* __restrict__ w_off,
                                const float* __restrict__ b_off,
                                const float* __restrict__ w_mod,
                                const float* __restrict__ b_mod,
                                float* __restrict__ offs,   // (B,18,H,W)
                                float* __restrict__ mods)   // (B,9,H,W)
{
    const int p = blockIdx.x * blockDim.x + threadIdx.x;   // pixel (b,y,x)
    const int j = blockIdx.y;                               // 0..26
    if (p >= Mtot) return;
    const int b   = p >> 12;
    const int rem = p & (PIX - 1);
    const int y   = rem >> 6;
    const int x   = rem & 63;

    const float* wbase = (j < 18) ? (w_off + j * Cc * 9)
                                  : (w_mod + (j - 18) * Cc * 9);
    float acc = (j < 18) ? b_off[j] : b_mod[j - 18];

    const float* qb = query + ((size_t)b * Cc << 12);
    for (int c = 0; c < Cc; ++c) {
        const float* qc = qb + ((size_t)c << 12);
        const float* wc = wbase + c * 9;
#pragma unroll
        for (int ky = 0; ky < 3; ++ky) {
            const int iy = y + ky - 1;
            if (iy < 0 || iy >= Hc) continue;
#pragma unroll
            for (int kx = 0; kx < 3; ++kx) {
                const int ix = x + kx - 1;
                if (ix < 0 || ix >= Wc) continue;
                acc = fmaf(qc[iy * Wc + ix], wc[ky * 3 + kx], acc);
            }
        }
    }
    if (j < 18) {
        offs[((size_t)(b * 18 + j) << 12) + rem] = acc;
    } else {
        mods[((size_t)(b * 9 + (j - 18)) << 12) + rem] = 1.0f / (1.0f + __expf(-acc));
    }
}

// ---------------------------------------------------------------------------
// Kernel 2: transpose w_out[o,c,n] -> w_t[k=c*9+n][o] (coalesced B loads later)
// ---------------------------------------------------------------------------
__global__ void dca_wt_kernel(const float* __restrict__ w_out,
                              float* __restrict__ w_t)
{
    const int t = blockIdx.x * blockDim.x + threadIdx.x;
    if (t >= Kk * OUTC) return;
    const int k = t / OUTC;          // c*9+n
    const int o = t - k * OUTC;      // consecutive across lanes -> coalesced store
    const int c = k / Nn;
    const int n = k - c * Nn;
    w_t[(size_t)k * OUTC + o] = w_out[((size_t)o * Cc + c) * Nn + n];
}

// ---------------------------------------------------------------------------
// Kernel 3: fused grid-sample gather + WMMA GEMM.
// Block = 256 threads (8 wave32), owns 32 consecutive pixels x 256 outc.
// grid.x = Mtot/32 = 512.
// ---------------------------------------------------------------------------
__global__ void __launch_bounds__(256)
dca_main_kernel(const float* __restrict__ value,
                const float* __restrict__ offs,
                const float* __restrict__ mods,
                const float* __restrict__ w_t,
                float* __restrict__ out)
{
    __shared__ float pw[32 * Nn * 4];        // bilinear weights (mod folded)
    __shared__ int   pidx[32 * Nn * 4];      // flat index into value plane
    __shared__ float a_lds[KROWS * ASTR];    // staged A chunk: [k-row][m]

    const int tid       = threadIdx.x;
    const int tile_base = blockIdx.x * 32;           // first pixel of tile
    const int b_tile    = tile_base >> 12;           // batch (constant per tile)

    // ---- setup: sampling indices + weights for 32 pixels x 9 points --------
    for (int s = tid; s < 32 * Nn; s += 256) {
        const int m = s / Nn;
        const int n = s - m * Nn;
        const int p   = tile_base + m;
        const int rem = p & (PIX - 1);
        const int y   = rem >> 6;
        const int x   = rem & 63;

        const float off_y = offs[((size_t)(b_tile * 18 + n)      << 12) + rem];
        const float off_x = offs[((size_t)(b_tile * 18 + 9 + n)  << 12) + rem];
        const float mval  = mods[((size_t)(b_tile * 9  + n)      << 12) + rem];

        const float py = off_y + (float)(n / 3 - 1) + (float)y;
        const float px = off_x + (float)(n % 3 - 1) + (float)x;
        // grid_sample align_corners=false on padded (66x66) image
        const float ys = py * (66.0f / 65.0f) - 0.5f + 1.0f;  // ... wait, see below
        const float xs = px * (66.0f / 65.0f) - 0.5f + 1.0f;
        // NOTE: the +1.0f above is wrong for the math; recompute cleanly:
        const float ysf = py * (66.0f / 65.0f) - 0.5f;
        const float xsf = px * (66.0f / 65.0f) - 0.5f;
        (void)ys; (void)xs;

        const float x0 = floorf(xsf);
        const float y0 = floorf(ysf);
        const float wx1 = xsf - x0, wx0 = 1.0f - wx1;
        const float wy1 = ysf - y0, wy0 = 1.0f - wy1;

#pragma unroll
        for (int jn = 0; jn < 4; ++jn) {
            const float xi = x0 + (float)(jn & 1);
            const float yi = y0 + (float)(jn >> 1);
            // padded coords valid AND maps into non-pad region: [1, 64]
            const bool ok = (xi >= 1.0f) && (xi <= 64.0f) &&
                            (yi >= 1.0f) && (yi <= 64.0f);
            float wgt = ((jn & 1) ? wx1 : wx0) * ((jn >> 1) ? wy1 : wy0) * mval;
            int idx = 0;
            if (ok) {
                idx = ((int)yi - 1) * Wc + ((int)xi - 1);
            } else {
                wgt = 0.0f;
            }
            pw  [(m * Nn + n) * 4 + jn] = wgt;
            pidx[(m * Nn + n) * 4 + jn] = idx;
        }
    }
    __syncthreads();

    // ---- wave tiling --------------------------------------------------------
    const int lane    = tid & 31;
    const int wave    = tid >> 5;          // 0..7
    const int moff    = (wave >> 2) * 16;  // waves 0-3 -> m 0..15, 4-7 -> 16..31
    const int colbase = (wave & 3) * 64;   // 4 N-tiles of 16 per wave
    const int hf      = lane >> 4;         // lane half (K interleave)
    const int ml      = lane & 15;

    v8f acc[4];
#pragma unroll
    for (int t = 0; t < 4; ++t) acc[t] = (v8f){0,0,0,0,0,0,0,0};

    const float* vbat = value + ((size_t)b_tile * Cc << 12);

    for (int c0 = 0; c0 < Cc; c0 += KC) {
        // ---- stage A: sample KC channels into LDS (2304 values) ------------
        for (int s = tid; s < KROWS * 32; s += 256) {
            const int m  = s & 31;
            const int r  = s >> 5;            // k-row in chunk = cc*9 + n
            const int cc = r / Nn;
            const int n  = r - cc * Nn;
            const float* vplane = vbat + ((size_t)(c0 + cc) << 12);
            const float* wq = &pw  [(m * Nn + n) * 4];
            const int*   iq = &pidx[(m * Nn + n) * 4];
            float v = wq[0] * vplane[iq[0]];
            v = fmaf(wq[1], vplane[iq[1]], v);
            v = fmaf(wq[2], vplane[iq[2]], v);
            v = fmaf(wq[3], vplane[iq[3]], v);
            a_lds[r * ASTR + m] = v;
        }
        __syncthreads();

        // ---- WMMA over this chunk: 18 k-steps of 4 -------------------------
        const int gkbase = c0 * Nn;           // global k of chunk start
#pragma unroll 2
        for (int kk = 0; kk < KROWS / 4; ++kk) {
            const int k0 = kk * 4;
            v2f a;
            a.x = a_lds[(k0 + 2 * hf + 0) * ASTR + moff + ml];
            a.y = a_lds[(k0 + 2 * hf + 1) * ASTR + moff + ml];
            const int gk = gkbase + k0 + 2 * hf;
            const float* wr0 = w_t + (size_t)gk * OUTC;
#pragma unroll
            for (int t = 0; t < 4; ++t) {
                const int col = colbase + t * 16 + ml;
                v2f bb;
                bb.x = wr0[col];
                bb.y = wr0[OUTC + col];
                acc[t] = __builtin_amdgcn_wmma_f32_16x16x4_f32(
                    false, a, false, bb, (short)0, acc[t], false, false);
            }
        }
        __syncthreads();
    }

    // ---- epilogue: D layout VGPR v -> M = v + 8*half, col = lane%16 --------
#pragma unroll
    for (int t = 0; t < 4; ++t) {
        const int o = colbase + t * 16 + ml;
        float* ob = out + ((size_t)(b_tile * OUTC + o) << 12);
#pragma unroll
        for (int v = 0; v < 8; ++v) {
            const int m = moff + v + 8 * hf;
            const int p = tile_base + m;
            ob[p & (PIX - 1)] = acc[t][v];
        }
    }
}

// ---------------------------------------------------------------------------
extern "C" void kernel_launch(void* const* d_in, const int* in_sizes, int n_in,
                              void* d_out, int out_size, void* d_ws, size_t ws_size,
                              hipStream_t stream) {
    const float* query = (const float*)d_in[0];
    const float* value = (const float*)d_in[1];
    const float* w_off = (const float*)d_in[2];
    const float* b_off = (const float*)d_in[3];
    const float* w_mod = (const float*)d_in[4];
    const float* b_mod = (const float*)d_in[5];
    const float* w_out = (const float*)d_in[6];
    float* out = (float*)d_out;

    float* offs = (float*)d_ws;                       // B*18*PIX
    float* mods = offs + (size_t)Bc * 18 * PIX;       // B*9*PIX
    float* w_t  = mods + (size_t)Bc * 9 * PIX;        // Kk*OUTC

    // offset + modulator convs
    dim3 cgrid(Mtot / 256, 27);
    dca_conv_kernel<<<cgrid, 256, 0, stream>>>(query, w_off, b_off, w_mod, b_mod,
                                               offs, mods);
    // weight transpose
    dca_wt_kernel<<<(Kk * OUTC + 255) / 256, 256, 0, stream>>>(w_out, w_t);
    // fused gather + WMMA GEMM
    dca_main_kernel<<<Mtot / 32, 256, 0, stream>>>(value, offs, mods, w_t, out);
}